// VQCodebook_74663711473893
// MI455X (gfx1250) — compile-verified
//
#include <hip/hip_runtime.h>
#include <hip/hip_bf16.h>

// VQ codebook lookup for MI455X (gfx1250, wave32, WMMA).
// score = codebook @ z_tile^T via 4x chained V_WMMA_F32_16X16X4_F32 (exact f32),
// accumulator seeded with -0.5*||e_k||^2 so argmax(score) == argmin(distance).

typedef __attribute__((ext_vector_type(2))) float v2f;
typedef __attribute__((ext_vector_type(8))) float v8f;

#define VQ_D 16
#define VQ_K 10
#define TILE_ROWS 16
#define WAVES_PER_BLOCK 8
#define ROWS_PER_BLOCK (TILE_ROWS * WAVES_PER_BLOCK)   // 128

__global__ void vq_init_kernel(float* __restrict__ loss) { *loss = 0.0f; }

__global__ void vq_final_kernel(float* __restrict__ loss, float inv_n) { *loss *= inv_n; }

__global__ __launch_bounds__(256) void vq_main_kernel(
    const float* __restrict__ z_e,      // (B, 16)
    const float* __restrict__ codebook, // (10, 16)
    float* __restrict__ z_q_out,        // (B, 16)
    float* __restrict__ idx_out,        // (B,) as float
    float* __restrict__ loss_out)       // scalar accumulator
{
    __shared__ __align__(16) float lds_cb[16 * 16];  // codebook zero-padded to 16x16
    __shared__ float lds_c0[16];                     // -0.5*||e_m||^2 (pad: -3e38)
    __shared__ float lds_lsum;

    const int tid  = threadIdx.x;
    const int wave = tid >> 5;
    const int lane = tid & 31;
    const int half = lane >> 4;   // 0: lanes 0-15, 1: lanes 16-31
    const int lrow = lane & 15;

    // Stage codebook into LDS (zero-pad codes 10..15); codebook read once per block.
    {
        const int m = tid >> 4, d = tid & 15;
        lds_cb[tid] = (m < VQ_K) ? codebook[m * VQ_D + d] : 0.0f;
    }
    if (tid == 0) lds_lsum = 0.0f;
    __syncthreads();
    if (tid < 16) {
        float s = 0.0f;
        #pragma unroll
        for (int d = 0; d < VQ_D; ++d) { float v = lds_cb[tid * VQ_D + d]; s += v * v; }
        lds_c0[tid] = (tid < VQ_K) ? (-0.5f * s) : -3.0e38f;
    }
    __syncthreads();

    const int row0  = (blockIdx.x * WAVES_PER_BLOCK + wave) * TILE_ROWS;
    const int myrow = row0 + lrow;

    // B-matrix (4x16 f32 per chunk): lanes 0-15 hold K={4c,4c+1}, lanes 16-31 K={4c+2,4c+3}.
    // Each lane: one 8B streaming load of z_e per chunk.
    v2f b[4];
    #pragma unroll
    for (int c = 0; c < 4; ++c) {
        const float* p = z_e + (size_t)myrow * VQ_D + 4 * c + 2 * half;
        b[c] = *(const v2f*)p;
    }
    // A-matrix (16x4 f32 per chunk): same lane pattern, rows = code index, from LDS.
    v2f a[4];
    #pragma unroll
    for (int c = 0; c < 4; ++c) {
        const float* p = &lds_cb[lrow * VQ_D + 4 * c + 2 * half];
        a[c] = *(const v2f*)p;
    }
    // C/D (16x16 f32): VGPR r holds M = r + 8*half, N = lrow. Seed with -0.5*||e||^2.
    v8f acc;
    #pragma unroll
    for (int r = 0; r < 8; ++r) acc[r] = lds_c0[8 * half + r];

    // score[M=code][N=row] over K=16, exact fp32, 4 chained WMMAs.
    acc = __builtin_amdgcn_wmma_f32_16x16x4_f32(false, a[0], false, b[0], (short)0, acc, false, false);
    acc = __builtin_amdgcn_wmma_f32_16x16x4_f32(false, a[1], false, b[1], (short)0, acc, false, false);
    acc = __builtin_amdgcn_wmma_f32_16x16x4_f32(false, a[2], false, b[2], (short)0, acc, false, false);
    acc = __builtin_amdgcn_wmma_f32_16x16x4_f32(false, a[3], false, b[3], (short)0, acc, false, false);

    // Per-lane argmax over the 8 codes this lane holds (keep lowest index on ties).
    float bestv = acc[0];
    int   besti = 8 * half;
    #pragma unroll
    for (int r = 1; r < 8; ++r) {
        if (acc[r] > bestv) { bestv = acc[r]; besti = 8 * half + r; }
    }
    // Combine the two half-lanes that hold the same column N.
    {
        const float ov = __shfl_xor(bestv, 16, 32);
        const int   oi = __shfl_xor(besti, 16, 32);
        if (ov > bestv || (ov == bestv && oi < besti)) { bestv = ov; besti = oi; }
    }

    // Gather + store z_q (= z_q_st forward value): each half-lane writes 8 dims (2x b128).
    {
        const float4 q0 = *(const float4*)&lds_cb[besti * VQ_D + 8 * half];
        const float4 q1 = *(const float4*)&lds_cb[besti * VQ_D + 8 * half + 4];
        float4* dst = (float4*)(z_q_out + (size_t)myrow * VQ_D + 8 * half);
        dst[0] = q0;
        dst[1] = q1;
    }
    if (half == 0) idx_out[myrow] = (float)besti;

    // Commitment-loss partial: this lane already holds z_e dims {4c+2*half, +1} in b[].
    float lsum = 0.0f;
    #pragma unroll
    for (int c = 0; c < 4; ++c) {
        const int d0 = 4 * c + 2 * half;
        const float da = b[c][0] - lds_cb[besti * VQ_D + d0];
        const float db = b[c][1] - lds_cb[besti * VQ_D + d0 + 1];
        lsum += da * da + db * db;
    }
    #pragma unroll
    for (int off = 16; off > 0; off >>= 1) lsum += __shfl_xor(lsum, off, 32);
    if (lane == 0) atomicAdd(&lds_lsum, lsum);
    __syncthreads();
    if (tid == 0) atomicAdd(loss_out, lds_lsum);
}

extern "C" void kernel_launch(void* const* d_in, const int* in_sizes, int n_in,
                              void* d_out, int out_size, void* d_ws, size_t ws_size,
                              hipStream_t stream) {
    const float* z_e = (const float*)d_in[0];
    const float* cb  = (const float*)d_in[1];
    const int Brows  = in_sizes[0] / VQ_D;   // 4194304

    float* out   = (float*)d_out;
    float* zq    = out;                              // B*16 floats
    float* idx   = out + (size_t)Brows * VQ_D;       // B floats
    float* loss  = idx + Brows;                      // 1 float

    vq_init_kernel<<<1, 1, 0, stream>>>(loss);

    const int blocks = Brows / ROWS_PER_BLOCK;       // 32768
    vq_main_kernel<<<blocks, 256, 0, stream>>>(z_e, cb, zq, idx, loss);

    vq_final_kernel<<<1, 1, 0, stream>>>(loss, 1.0f / ((float)Brows * (float)VQ_D));
}